// FruitFly_54795192762755
// MI455X (gfx1250) — compile-verified
//
#include <hip/hip_runtime.h>
#include <stdint.h>

#define KROWS 2048
#define NVOCAB 50000
#define NBATCH 4096
#define WN 11
#define NIDS (NBATCH * WN)          // 45056
#define CHUNK 25000                 // floats per staged chunk (100000 B), 2 chunks/row
#define B128_PER_CHUNK (CHUNK / 4)  // 6250
#define BLK 256
#define BPT (NBATCH / BLK)          // 16 batch elements per thread
#define ROWS_PER_WG 4
#define NWG (KROWS / ROWS_PER_WG)   // 512 persistent workgroups
#define NTOT (ROWS_PER_WG * 2)      // 8 chunks per workgroup
// floor(6250/256) = 24: with in-order async completion, waiting cnt<=24 proves
// every op older than the newest 24 (i.e. all of the previous chunk) is done.
#define WAIT_PREF 24

// ---------- gfx1250 async global->LDS copy (ASYNCcnt path) ----------
__device__ __forceinline__ void async_g2l_b128(uint32_t lds_byte, uint64_t gaddr) {
  asm volatile("global_load_async_to_lds_b128 %0, %1, off"
               :: "v"(lds_byte), "v"(gaddr) : "memory");
}
__device__ __forceinline__ void wait_async_pref() {
  asm volatile("s_wait_asynccnt 24" ::: "memory");
}
__device__ __forceinline__ void wait_async_all() {
  asm volatile("s_wait_asynccnt 0" ::: "memory");
}

// order-preserving float -> u32 map (monotonic incl. negatives)
__device__ __forceinline__ unsigned ord_of_float(float f) {
  unsigned u = __float_as_uint(f);
  return (u & 0x80000000u) ? ~u : (u | 0x80000000u);
}

// ------------------------------ pass 1 ------------------------------
// Persistent WG handles ROWS_PER_WG contiguous rows. Row data streams
// HBM -> LDS via double-buffered async b128 copies (prefetch chunk t+1
// while computing chunk t). ids live in LDS as 12 padded u16 per batch
// element (3x ds_load_b64 per gather pass, slot 12 = 0xFFFF sentinel).
// Per-b argmax is carried in registers; one packed atomicMax per b at end.
__global__ void __launch_bounds__(BLK)
fruitfly_pass1(const float* __restrict__ W, const int* __restrict__ ids,
               unsigned long long* __restrict__ packed,
               float* __restrict__ norms2) {
  __shared__ __align__(16) float s_buf[2][CHUNK];      // 200000 B
  __shared__ float s_red[BLK];                         // 1024 B
  __shared__ unsigned short s_idsp[NBATCH * 12];       // 98304 B

  const int tid = threadIdx.x;
  const int krow0 = blockIdx.x * ROWS_PER_WG;
  const uint64_t wbase = (uint64_t)(uintptr_t)(W + (size_t)krow0 * NVOCAB);
  const uint32_t buf_lds0 = (uint32_t)(uintptr_t)&s_buf[0][0];
  const uint32_t buf_lds1 = (uint32_t)(uintptr_t)&s_buf[1][0];

  // ---- kick off chunk 0 before anything else (hide first latency) ----
  {
    const uint64_t src = wbase;
    for (int i = tid; i < B128_PER_CHUNK; i += BLK)
      async_g2l_b128(buf_lds0 + (uint32_t)i * 16u, src + (uint64_t)i * 16u);
  }

  // ---- stage padded ids (overlaps with chunk-0 flight) ----
  for (int b = tid; b < NBATCH; b += BLK) {
    const int* ip = ids + b * WN;
    unsigned* dst = (unsigned*)(s_idsp + b * 12);
#pragma unroll
    for (int j = 0; j < 5; ++j)
      dst[j] = ((unsigned)ip[2 * j] & 0xFFFFu) | (((unsigned)ip[2 * j + 1] & 0xFFFFu) << 16);
    dst[5] = ((unsigned)ip[10] & 0xFFFFu) | 0xFFFF0000u;  // sentinel id
  }

  float acts[BPT];
  unsigned long long best[BPT];
#pragma unroll
  for (int j = 0; j < BPT; ++j) { acts[j] = 0.0f; best[j] = 0ull; }
  float nrm = 0.0f;

  for (int t = 0; t < NTOT; ++t) {
    // ---- prefetch chunk t+1 into the other buffer ----
    if (t < NTOT - 1) {
      const int tn = t + 1;
      const uint64_t src = wbase + (uint64_t)(tn >> 1) * (NVOCAB * 4u)
                                 + (uint64_t)(tn & 1) * (CHUNK * 4u);
      const uint32_t dst = (tn & 1) ? buf_lds1 : buf_lds0;
      for (int i = tid; i < B128_PER_CHUNK; i += BLK)
        async_g2l_b128(dst + (uint32_t)i * 16u, src + (uint64_t)i * 16u);
      wait_async_pref();   // chunk t fully landed; t+1 stays in flight
    } else {
      wait_async_all();
    }
    __syncthreads();       // chunk t (and, on t==0, s_idsp) visible to all waves

    const float* buf = s_buf[t & 1];
    const int base = (t & 1) * CHUNK;

    // ---- row-norm^2 partial: conflict-free float4 sweep ----
    {
      const float4* b4 = (const float4*)buf;
      for (int i = tid; i < B128_PER_CHUNK; i += BLK) {
        float4 v = b4[i];
        nrm += v.x * v.x + v.y * v.y + v.z * v.z + v.w * v.w;
      }
    }

    // ---- gather-accumulate acts (branchless, clamped LDS index) ----
#pragma unroll
    for (int j = 0; j < BPT; ++j) {
      const int b = tid + j * BLK;
      const uint2* idp = (const uint2*)(s_idsp + b * 12);  // 3x ds_load_b64
      uint2 p0 = idp[0], p1 = idp[1], p2 = idp[2];
      unsigned words[6] = {p0.x, p0.y, p1.x, p1.y, p2.x, p2.y};
      float a = acts[j];
#pragma unroll
      for (int q = 0; q < 6; ++q) {
#pragma unroll
        for (int h = 0; h < 2; ++h) {
          const unsigned id = (h == 0) ? (words[q] & 0xFFFFu) : (words[q] >> 16);
          const unsigned uv = (unsigned)((int)id - base);
          const unsigned idxc = (uv < (unsigned)CHUNK) ? uv : 0u;
          const float val = buf[idxc];
          a += (uv < (unsigned)CHUNK) ? val : 0.0f;
        }
      }
      acts[j] = a;
    }
    __syncthreads();  // all reads of buf[t&1] done before it is re-staged

    // ---- end of a row: finalize norm + fold into register argmax ----
    if (t & 1) {
      const int k = krow0 + (t >> 1);
      s_red[tid] = nrm;
      __syncthreads();
      for (int s = BLK / 2; s > 0; s >>= 1) {
        if (tid < s) s_red[tid] += s_red[tid + s];
        __syncthreads();
      }
      if (tid == 0) norms2[k] = s_red[0];
      nrm = 0.0f;

      const unsigned kcode = (unsigned)(KROWS - 1 - k);  // ties -> smaller k wins
#pragma unroll
      for (int j = 0; j < BPT; ++j) {
        const unsigned long long p =
            ((unsigned long long)ord_of_float(acts[j]) << 32) | kcode;
        if (p > best[j]) best[j] = p;
        acts[j] = 0.0f;
      }
    }
  }

  // ---- one packed atomic max per owned batch element ----
#pragma unroll
  for (int j = 0; j < BPT; ++j)
    atomicMax(&packed[tid + j * BLK], best[j]);
}

// ------------------------------ pass 2 ------------------------------
__global__ void __launch_bounds__(BLK)
fruitfly_pass2(const float* __restrict__ W, const int* __restrict__ ids,
               const float* __restrict__ Ps,
               const unsigned long long* __restrict__ packed,
               const float* __restrict__ norms2, float* __restrict__ out) {
  __shared__ float s_red[BLK];
  const int b = blockIdx.x * BLK + threadIdx.x;
  float contrib = 0.0f;
  if (b < NBATCH) {
    const unsigned low = (unsigned)(packed[b] & 0xFFFFFFFFull);
    const int mu = (KROWS - 1) - (int)low;
    const float* wrow = W + (size_t)mu * NVOCAB;
    const int* ip = ids + b * WN;
    const float* pp = Ps + b * WN;
    float num = 0.0f;
#pragma unroll
    for (int w = 0; w < WN; ++w) num += pp[w] * wrow[ip[w]];
    contrib = num / sqrtf(norms2[mu]);
  }
  s_red[threadIdx.x] = contrib;
  __syncthreads();
  for (int s = BLK / 2; s > 0; s >>= 1) {
    if (threadIdx.x < s) s_red[threadIdx.x] += s_red[threadIdx.x + s];
    __syncthreads();
  }
  if (threadIdx.x == 0) atomicAdd(out, -s_red[0]);
}

// ------------------------------ init ------------------------------
__global__ void fruitfly_init(unsigned long long* packed, float* out) {
  const int i = blockIdx.x * blockDim.x + threadIdx.x;
  if (i < NBATCH) packed[i] = 0ull;
  if (i == 0) out[0] = 0.0f;
}

// ---------------------------- launcher ----------------------------
extern "C" void kernel_launch(void* const* d_in, const int* in_sizes, int n_in,
                              void* d_out, int out_size, void* d_ws, size_t ws_size,
                              hipStream_t stream) {
  (void)in_sizes; (void)n_in; (void)out_size; (void)ws_size;
  const int* ids  = (const int*)d_in[0];    // [B*Wn] int32
  const float* Ps = (const float*)d_in[1];  // [B, Wn] f32
  // d_in[2] = pos, d_in[3] = top_k : unused by the reference
  const float* W  = (const float*)d_in[4];  // [K, V] f32

  float* out = (float*)d_out;
  unsigned long long* packed = (unsigned long long*)d_ws;          // 32 KB
  float* norms2 = (float*)((char*)d_ws + NBATCH * sizeof(unsigned long long)); // 8 KB

  fruitfly_init<<<dim3((NBATCH + 255) / 256), dim3(256), 0, stream>>>(packed, out);
  fruitfly_pass1<<<dim3(NWG), dim3(BLK), 0, stream>>>(W, ids, packed, norms2);
  fruitfly_pass2<<<dim3(NBATCH / BLK), dim3(BLK), 0, stream>>>(W, ids, Ps, packed, norms2, out);
}